// ExponentialSmoothedCCCLoss_64759516889966
// MI455X (gfx1250) — compile-verified
//
#include <hip/hip_runtime.h>

typedef float v2f __attribute__((ext_vector_type(2)));
typedef float v8f __attribute__((ext_vector_type(8)));

#define ALPHA_F 0.9f
#define BETA_D  0.5
#define KWIN    16        // 0.1^16 = 1e-16 relative: exact at fp32
#define NBLOCKS 1024
#define TPB     256       // 8 wave32 per block

// decay^l by exact repeated multiply (setup / one-off only)
__device__ __forceinline__ float pow_decay(int l, float decay) {
    float r = 1.0f;
    for (int i = 0; i < l; ++i) r *= decay;
    return r;
}

// Slow-path tile: clamped+zero-padded loads, WMMA chain, tile-0 boundary
// correction, bounds-guarded accumulation. Called outside the hot loop for
// tile 0 and a ragged tail tile only (wave-uniform caller -> EXEC all ones).
__device__ __forceinline__ void process_tile_slow(
    const float* __restrict__ yt, const float* __restrict__ yp,
    int tile, int T, int n, int half, float decay,
    const float wa0[8], const float wa1[8],
    float& s1t, float& s2t, float& s1p, float& s2p, float& sxy) {
    const int base = tile * 256;
    const int lane_base = base + 16 * (n - 1) + 2 * half;   // even
    int qmax = (T - 2) & ~1;
    if (qmax < 0) qmax = 0;

    v2f bt[8], bp[8];
#pragma unroll
    for (int c = 0; c < 8; ++c) {
        int q0 = lane_base + 4 * c;                         // even
        int q0c = q0 < 0 ? 0 : (q0 > qmax ? qmax : q0);
        v2f vt = *(const v2f*)(yt + q0c);
        v2f vp = *(const v2f*)(yp + q0c);
        bt[c] = (q0 < 0) ? (v2f){0.0f, 0.0f} : vt;
        bp[c] = (q0 < 0) ? (v2f){0.0f, 0.0f} : vp;
    }

    v8f ct = {}; v8f cp = {};
#pragma unroll
    for (int c = 0; c < 8; ++c) {
        v2f a = {wa0[c], wa1[c]};
        ct = __builtin_amdgcn_wmma_f32_16x16x4_f32(
            false, a, false, bt[c], (short)0, ct, false, false);
        cp = __builtin_amdgcn_wmma_f32_16x16x4_f32(
            false, a, false, bp[c], (short)0, cp, false, false);
    }

    // Exact boundary: s[i] (i<16) needs extra decay^(i+1)*x[0]
    // (zero-padded window yields alpha*decay^i*x[0]; exact recurrence
    //  carries decay^i*x[0]; difference = decay^(i+1)*x[0]).
    if (tile == 0 && n == 0) {
        float x0t = yt[0], x0p = yp[0];
#pragma unroll
        for (int r = 0; r < 8; ++r) {
            int i = r + 8 * half;                 // global index (n==0)
            float w = pow_decay(i + 1, decay);
            ct[r] += w * x0t;
            cp[r] += w * x0p;
        }
    }

    // D layout: VGPR r, lane -> element i = base + 16n + r + 8*half
#pragma unroll
    for (int r = 0; r < 8; ++r) {
        int i = base + 16 * n + r + 8 * half;
        if (i < T) {
            float a = ct[r], b = cp[r];
            s1t += a; s2t += a * a;
            s1p += b; s2p += b * b;
            sxy += a * b;
        }
    }
}

// Fused windowed-EMA (WMMA Toeplitz matmul) + 5-way statistics partials.
// Band padded to 17 columns (lag l = m+16-p', active p' = m+1..m+16) so every
// per-lane B pair sits at an EVEN element offset -> aligned b64 loads with
// immediate offsets. Hot loop handles only fully-interior tiles: branch-free.
__global__ void __launch_bounds__(TPB) ema_ccc_partial(
    const float* __restrict__ yt, const float* __restrict__ yp,
    double* __restrict__ part, int T) {
    const float decay = 1.0f - ALPHA_F;

    const int lane  = threadIdx.x & 31;
    const int wave  = threadIdx.x >> 5;
    const int n     = lane & 15;      // B column / D column / A row (m = n)
    const int half  = lane >> 4;      // K-slice selector (K = 2*half, 2*half+1)
    const int total_waves = gridDim.x * (TPB / 32);
    const int wave_gid    = blockIdx.x * (TPB / 32) + wave;

    const int NT      = (T + 255) >> 8;   // all tiles
    const int NT_full = T >> 8;           // tiles with base+256 <= T

    // A operand (banded Toeplitz weights), per-lane constants:
    // chunk c supplies columns p' = 4c+2*half, +1 of row m=n;
    // weight = alpha*decay^l for l = m+16-p' in [0,15], else 0.
    float wa0[8], wa1[8];
#pragma unroll
    for (int c = 0; c < 8; ++c) {
        int p0 = 4 * c + 2 * half;
        int l0 = n + 16 - p0;
        int l1 = l0 - 1;
        wa0[c] = (l0 >= 0 && l0 < KWIN) ? ALPHA_F * pow_decay(l0, decay) : 0.0f;
        wa1[c] = (l1 >= 0 && l1 < KWIN) ? ALPHA_F * pow_decay(l1, decay) : 0.0f;
    }

    float s1t = 0.f, s2t = 0.f, s1p = 0.f, s2p = 0.f, sxy = 0.f;

    // ---- hot loop: interior tiles only, no conditionals ----
    const int lane_off0 = 16 * (n - 1) + 2 * half;   // even
    for (int tile = 1 + wave_gid; tile < NT_full; tile += total_waves) {
        const int lane_base = tile * 256 + lane_off0;
        const v2f* pt2 = (const v2f*)(yt + lane_base);
        const v2f* pp2 = (const v2f*)(yp + lane_base);

        v2f bt[8], bp[8];
#pragma unroll
        for (int c = 0; c < 8; ++c) {   // imm offsets 0,16,...,112 bytes
            bt[c] = pt2[2 * c];
            bp[c] = pp2[2 * c];
        }

        v8f ct = {}; v8f cp = {};
#pragma unroll
        for (int c = 0; c < 8; ++c) {
            v2f a = {wa0[c], wa1[c]};
            ct = __builtin_amdgcn_wmma_f32_16x16x4_f32(
                false, a, false, bt[c], (short)0, ct, false, false);
            cp = __builtin_amdgcn_wmma_f32_16x16x4_f32(
                false, a, false, bp[c], (short)0, cp, false, false);
        }

#pragma unroll
        for (int r = 0; r < 8; ++r) {
            float a = ct[r], b = cp[r];
            s1t += a; s2t += a * a;
            s1p += b; s2p += b * b;
            sxy += a * b;
        }
    }

    // ---- peeled special tiles (once, by global wave 0) ----
    if (wave_gid == 0) {
        process_tile_slow(yt, yp, 0, T, n, half, decay, wa0, wa1,
                          s1t, s2t, s1p, s2p, sxy);
        if (NT > NT_full && NT_full > 0)   // ragged tail tile
            process_tile_slow(yt, yp, NT_full, T, n, half, decay, wa0, wa1,
                              s1t, s2t, s1p, s2p, sxy);
    }

    // wave32 reduction (deterministic tree)
#pragma unroll
    for (int off = 16; off > 0; off >>= 1) {
        s1t += __shfl_down(s1t, off, 32);
        s2t += __shfl_down(s2t, off, 32);
        s1p += __shfl_down(s1p, off, 32);
        s2p += __shfl_down(s2p, off, 32);
        sxy += __shfl_down(sxy, off, 32);
    }

    __shared__ double sm[TPB / 32][5];
    if (lane == 0) {
        sm[wave][0] = (double)s1t; sm[wave][1] = (double)s2t;
        sm[wave][2] = (double)s1p; sm[wave][3] = (double)s2p;
        sm[wave][4] = (double)sxy;
    }
    __syncthreads();
    if (threadIdx.x == 0) {
        double a0 = 0, a1 = 0, a2 = 0, a3 = 0, a4 = 0;
        for (int w = 0; w < TPB / 32; ++w) {
            a0 += sm[w][0]; a1 += sm[w][1]; a2 += sm[w][2];
            a3 += sm[w][3]; a4 += sm[w][4];
        }
        double* o = part + (size_t)blockIdx.x * 5;
        o[0] = a0; o[1] = a1; o[2] = a2; o[3] = a3; o[4] = a4;
    }
}

__global__ void __launch_bounds__(TPB) ccc_final(
    const double* __restrict__ part, float* __restrict__ out, int T) {
    double s0 = 0, s1 = 0, s2 = 0, s3 = 0, s4 = 0;
    for (int b = threadIdx.x; b < NBLOCKS; b += TPB) {
        const double* p = part + (size_t)b * 5;
        s0 += p[0]; s1 += p[1]; s2 += p[2]; s3 += p[3]; s4 += p[4];
    }
#pragma unroll
    for (int off = 16; off > 0; off >>= 1) {
        s0 += __shfl_down(s0, off, 32);
        s1 += __shfl_down(s1, off, 32);
        s2 += __shfl_down(s2, off, 32);
        s3 += __shfl_down(s3, off, 32);
        s4 += __shfl_down(s4, off, 32);
    }
    __shared__ double sm[TPB / 32][5];
    int wave = threadIdx.x >> 5, lane = threadIdx.x & 31;
    if (lane == 0) {
        sm[wave][0] = s0; sm[wave][1] = s1; sm[wave][2] = s2;
        sm[wave][3] = s3; sm[wave][4] = s4;
    }
    __syncthreads();
    if (threadIdx.x == 0) {
        double S1t = 0, S2t = 0, S1p = 0, S2p = 0, Sxy = 0;
        for (int w = 0; w < TPB / 32; ++w) {
            S1t += sm[w][0]; S2t += sm[w][1]; S1p += sm[w][2];
            S2p += sm[w][3]; Sxy += sm[w][4];
        }
        double Td   = (double)T;
        double mt   = S1t / Td, mp = S1p / Td;
        double vart = (S2t - S1t * S1t / Td) / (Td - 1.0);   // ddof=1
        double varp = (S2p - S1p * S1p / Td) / (Td - 1.0);
        double cov0 = (Sxy - S1t * S1p / Td) / Td;            // ddof=0
        // numerator 2*cor*sd_t*sd_p == 2*cov0 (sd's cancel)
        double denom = vart + varp + (mt - mp) * (mt - mp);
        double ccc   = 2.0 * cov0 / denom;
        out[0] = (float)((1.0 - ccc) * BETA_D);
    }
}

extern "C" void kernel_launch(void* const* d_in, const int* in_sizes, int n_in,
                              void* d_out, int out_size, void* d_ws, size_t ws_size,
                              hipStream_t stream) {
    const float* yt = (const float*)d_in[0];
    const float* yp = (const float*)d_in[1];
    const int T = in_sizes[0];
    double* part = (double*)d_ws;   // NBLOCKS * 5 doubles = 40 KB

    ema_ccc_partial<<<NBLOCKS, TPB, 0, stream>>>(yt, yp, part, T);
    ccc_final<<<1, TPB, 0, stream>>>(part, (float*)d_out, T);
}